// ProductGNN_88115549044789
// MI455X (gfx1250) — compile-verified
//
#include <hip/hip_runtime.h>

#define N_NODES 100000
#define N_EDGES 1200000
#define DIM 64
#define E_TOT (N_EDGES + N_NODES)   // edges + self loops

typedef __attribute__((ext_vector_type(2))) float v2f;
typedef __attribute__((ext_vector_type(8))) float v8f;

// ---------------------------------------------------------------------------
// Degree / normalization
// ---------------------------------------------------------------------------
__global__ void deg_init_kernel(float* __restrict__ deg) {
    int i = blockIdx.x * blockDim.x + threadIdx.x;
    if (i < N_NODES) deg[i] = 1.0f;              // self-loop contribution
}

__global__ void deg_accum_kernel(const int* __restrict__ ei, float* __restrict__ deg) {
    int e = blockIdx.x * blockDim.x + threadIdx.x;
    if (e < N_EDGES) atomicAdd(&deg[ei[N_EDGES + e]], 1.0f);   // dst row of edge_index
}

__global__ void dinv_kernel(float* __restrict__ deg) {
    int i = blockIdx.x * blockDim.x + threadIdx.x;
    if (i < N_NODES) deg[i] = rsqrtf(deg[i]);    // deg >= 1 always
}

// ---------------------------------------------------------------------------
// GEMM: out[N,64] = in[N,64] @ W[64,64] using V_WMMA_F32_16X16X4_F32.
// One wave owns a fixed 16-wide column tile (col0) and loops over row tiles,
// so the 16 B-fragments of W are loaded once per wave and held in VGPRs.
// ---------------------------------------------------------------------------
__global__ void gemm_wmma_kernel(const float* __restrict__ in,
                                 const float* __restrict__ W,
                                 float* __restrict__ out) {
    const int lane = threadIdx.x & 31;
    const int wid  = (blockIdx.x * blockDim.x + threadIdx.x) >> 5;
    const int nWaves = (gridDim.x * blockDim.x) >> 5;

    const int col0 = (wid & 3) * 16;     // 4 column tiles cover DIM=64
    const int m    = lane & 15;          // A-matrix row within tile / B,C col
    const int hi   = lane >> 4;          // half-wave select
    const int klo  = hi * 2;             // K offset {0,2} per ISA 16x4 A layout

    // Preload B fragments: B[k][n] 4x16 per k-block; VGPR0 holds K={klo},
    // VGPR1 holds K={klo+1} (rows 0/2 then 1/3 split across half-waves).
    v2f bfrag[16];
#pragma unroll
    for (int kb = 0; kb < 16; ++kb) {
        int k = kb * 4 + klo;
        bfrag[kb].x = W[k * DIM + col0 + m];
        bfrag[kb].y = W[(k + 1) * DIM + col0 + m];
    }

    const int nRowTiles  = N_NODES / 16;       // 6250, exact
    const int rowStride  = nWaves >> 2;        // waves per column tile

    for (int rt = wid >> 2; rt < nRowTiles; rt += rowStride) {
        const int row0 = rt * 16;
        const float* arow = in + (size_t)(row0 + m) * DIM;
        v8f c = {};
#pragma unroll
        for (int kb = 0; kb < 16; ++kb) {
            v2f a = *(const v2f*)(arow + kb * 4 + klo);   // K=klo..klo+1 of block
            c = __builtin_amdgcn_wmma_f32_16x16x4_f32(
                    /*neg_a=*/false, a, /*neg_b=*/false, bfrag[kb],
                    /*c_mod=*/(short)0, c, /*reuse_a=*/false, /*reuse_b=*/false);
        }
        // C/D layout: VGPR r -> row (r + hi*8), col m
        float* orow = out + (size_t)(row0 + hi * 8) * DIM + col0 + m;
#pragma unroll
        for (int r = 0; r < 8; ++r) orow[(size_t)r * DIM] = c[r];
    }
}

// ---------------------------------------------------------------------------
// out[n][d] = bias[d]
// ---------------------------------------------------------------------------
__global__ void bias_fill_kernel(float* __restrict__ h, const float* __restrict__ b) {
    int i = blockIdx.x * blockDim.x + threadIdx.x;
    if (i < N_NODES * DIM) h[i] = b[i & (DIM - 1)];
}

// ---------------------------------------------------------------------------
// Edge scatter: one wave per edge, each lane carries 2 of the 64 dims.
// h[dst] += t[src] * (dinv[src]*dinv[dst]) via global_atomic_add_f32.
// Edges [0,E) come from edge_index; [E, E+N) are self loops.
// ---------------------------------------------------------------------------
__global__ void scatter_kernel(const float* __restrict__ t,
                               const int* __restrict__ ei,
                               const float* __restrict__ dinv,
                               float* __restrict__ h) {
    int tid = blockIdx.x * blockDim.x + threadIdx.x;
    int e = tid >> 5;
    int p = tid & 31;
    if (e >= E_TOT) return;
    int s, d;
    if (e < N_EDGES) { s = ei[e]; d = ei[N_EDGES + e]; }
    else             { s = e - N_EDGES; d = s; }
    float nm = dinv[s] * dinv[d];
    const float* mp = t + (size_t)s * DIM + p * 2;
    v2f msg = *(const v2f*)mp;
    float* op = h + (size_t)d * DIM + p * 2;
    atomicAdd(op,     msg.x * nm);
    atomicAdd(op + 1, msg.y * nm);
}

__global__ void relu_kernel(float* __restrict__ h) {
    int i = blockIdx.x * blockDim.x + threadIdx.x;
    if (i < N_NODES * DIM) h[i] = fmaxf(h[i], 0.0f);
}

// ---------------------------------------------------------------------------
extern "C" void kernel_launch(void* const* d_in, const int* in_sizes, int n_in,
                              void* d_out, int out_size, void* d_ws, size_t ws_size,
                              hipStream_t stream) {
    const float* x  = (const float*)d_in[0];
    const int*   ei = (const int*)d_in[1];
    const float* W1 = (const float*)d_in[2];
    const float* b1 = (const float*)d_in[3];
    const float* W2 = (const float*)d_in[4];
    const float* b2 = (const float*)d_in[5];
    const float* W3 = (const float*)d_in[6];
    const float* b3 = (const float*)d_in[7];
    float* out = (float*)d_out;

    char* ws = (char*)d_ws;
    size_t off = 0;
    float* dinv = (float*)(ws + off); off += ((size_t)N_NODES * 4 + 255) & ~(size_t)255;
    float* tbuf = (float*)(ws + off); off += (size_t)N_NODES * DIM * 4;
    float* hbuf = (float*)(ws + off);

    const int T = 256;
    const int gN   = (N_NODES + T - 1) / T;
    const int gE   = (N_EDGES + T - 1) / T;
    const int gND  = (N_NODES * DIM + T - 1) / T;
    const int gSc  = ((E_TOT * 32) + T - 1) / T;      // 162500 blocks
    const int gGemm = 512;                            // 4096 waves, 1024 per col tile

    // degree -> dinv (dinv buffer doubles as deg accumulator)
    deg_init_kernel<<<gN, T, 0, stream>>>(dinv);
    deg_accum_kernel<<<gE, T, 0, stream>>>(ei, dinv);
    dinv_kernel<<<gN, T, 0, stream>>>(dinv);

    // Layer 1: h = relu(scatter(x@W1) + b1)
    gemm_wmma_kernel<<<gGemm, T, 0, stream>>>(x, W1, tbuf);
    bias_fill_kernel<<<gND, T, 0, stream>>>(hbuf, b1);
    scatter_kernel<<<gSc, T, 0, stream>>>(tbuf, ei, dinv, hbuf);
    relu_kernel<<<gND, T, 0, stream>>>(hbuf);

    // Layer 2: h = relu(scatter(h@W2) + b2)
    gemm_wmma_kernel<<<gGemm, T, 0, stream>>>(hbuf, W2, tbuf);
    bias_fill_kernel<<<gND, T, 0, stream>>>(hbuf, b2);
    scatter_kernel<<<gSc, T, 0, stream>>>(tbuf, ei, dinv, hbuf);
    relu_kernel<<<gND, T, 0, stream>>>(hbuf);

    // Layer 3: out = scatter(h@W3) + b3
    gemm_wmma_kernel<<<gGemm, T, 0, stream>>>(hbuf, W3, tbuf);
    bias_fill_kernel<<<gND, T, 0, stream>>>(out, b3);
    scatter_kernel<<<gSc, T, 0, stream>>>(tbuf, ei, dinv, out);
}